// IWMaxSquareloss_20512763806262
// MI455X (gfx1250) — compile-verified
//
#include <hip/hip_runtime.h>
#include <stdint.h>

// Problem constants (from reference: inputs (8, 19, 512, 1024) fp32)
#define N_IMG    8
#define C_CLS    19
#define H_DIM    512
#define W_DIM    1024
#define HW       (H_DIM * W_DIM)        // 524288, class stride in elements
#define TILE_PIX 256                    // pixels per TDM tile (per block iter)
#define ITERS    8
#define CHUNK    (TILE_PIX * ITERS)     // 2048 pixels per block
#define BPG      (HW / CHUNK)           // 256 blocks per image
#define GRID     (N_IMG * BPG)          // 2048 blocks
#define RATIO_F  0.2f

typedef int v4i __attribute__((ext_vector_type(4)));
typedef int v8i __attribute__((ext_vector_type(8)));

// ---- CDNA5 TDM: tensor_load_to_lds with a 2D strided-tile descriptor ----
// Tile: tile_dim0 = TILE_PIX contiguous fp32 pixels, tile_dim1 = C_CLS rows,
// row stride = HW elements (2 MB apart in memory -> contiguous in LDS).
__device__ __forceinline__ void tdm_issue_tile(uint32_t lds_addr, uint64_t gaddr) {
    v4i g0;
    g0.x = 1;                                                   // count=1 (valid), user mode
    g0.y = (int)lds_addr;                                       // D#.lds_addr
    g0.z = (int)(uint32_t)gaddr;                                // global_addr[31:0]
    g0.w = (int)(((uint32_t)(gaddr >> 32) & 0x01FFFFFFu)        // global_addr[56:32]
                 | 0x80000000u);                                // type=2 ("image")
    v8i g1;
    g1.s0 = 0x00020000;                                         // wg_mask=0, data_size=2 (4B)
    g1.s1 = (int)((HW & 0xFFFFu) << 16);                        // tensor_dim0[15:0] (=0 here)
    g1.s2 = (int)((HW >> 16) | ((C_CLS & 0xFFFFu) << 16));      // tensor_dim0[31:16] | tensor_dim1[15:0]
    g1.s3 = (int)(TILE_PIX << 16);                              // tensor_dim1[31:16]=0 | tile_dim0
    g1.s4 = C_CLS;                                              // tile_dim1 | tile_dim2=0
    g1.s5 = HW;                                                 // tensor_dim0_stride[31:0]
    g1.s6 = 0;                                                  // stride0[47:32] | stride1[15:0]
    g1.s7 = 0;                                                  // stride1[47:16]
    // 2D tensor: VADDR2/VADDR3 = NULL (2-group form)
    asm volatile("tensor_load_to_lds %0, %1" :: "s"(g0), "s"(g1) : "memory");
}

__device__ __forceinline__ void wait_tensorcnt_le1() {
#if __has_builtin(__builtin_amdgcn_s_wait_tensorcnt)
    __builtin_amdgcn_s_wait_tensorcnt(1);
#else
    asm volatile("s_wait_tensorcnt 0x1" ::: "memory");
#endif
}
__device__ __forceinline__ void wait_tensorcnt_le0() {
#if __has_builtin(__builtin_amdgcn_s_wait_tensorcnt)
    __builtin_amdgcn_s_wait_tensorcnt(0);
#else
    asm volatile("s_wait_tensorcnt 0x0" ::: "memory");
#endif
}

// ---------------- Kernel 0: zero the global class histogram ----------------
__global__ void iw_init(unsigned* __restrict__ gHist) {
    int t = threadIdx.x;
    if (t < N_IMG * C_CLS) gHist[t] = 0u;
}

// ---------------- Kernel 1: softmax + Σp² per class + argmax histogram -----
__global__ void __launch_bounds__(256)
iw_main(const float* __restrict__ in, float* __restrict__ Spart,
        unsigned* __restrict__ gHist) {
    __shared__ float tile[2][C_CLS * TILE_PIX];   // double-buffered TDM tiles
    __shared__ float wsum[8 * C_CLS];             // per-wave partial sums
    __shared__ unsigned histS[C_CLS];             // block-local histogram

    const int tid   = threadIdx.x;
    const int n     = blockIdx.x / BPG;           // image index
    const int chunk = blockIdx.x % BPG;           // pixel-chunk within image
    const uint64_t gbase = (uint64_t)(uintptr_t)in
        + ((uint64_t)n * C_CLS * HW + (uint64_t)chunk * CHUNK) * 4ull;

    if (tid < C_CLS) histS[tid] = 0u;

    const uint32_t lds0 = (uint32_t)(uintptr_t)&tile[0][0];
    const uint32_t lds1 = (uint32_t)(uintptr_t)&tile[1][0];

    float acc[C_CLS];
#pragma unroll
    for (int c = 0; c < C_CLS; ++c) acc[c] = 0.0f;

    const bool issuer = (tid < 32);               // wave 0 issues tensor ops
    if (issuer) tdm_issue_tile(lds0, gbase);      // prologue: tile 0

    for (int it = 0; it < ITERS; ++it) {
        if (issuer) {
            if (it + 1 < ITERS) {                 // prefetch next tile
                tdm_issue_tile((it & 1) ? lds0 : lds1,
                               gbase + (uint64_t)(it + 1) * TILE_PIX * 4ull);
                wait_tensorcnt_le1();             // oldest (current) tile done
            } else {
                wait_tensorcnt_le0();             // last tile done
            }
        }
        __syncthreads();                          // tile visible to all waves

        const float* tp = (it & 1) ? &tile[1][0] : &tile[0][0];
        float x[C_CLS];
        float m = -3.402823466e38f;
        int am = 0;
#pragma unroll
        for (int c = 0; c < C_CLS; ++c) {         // conflict-free: stride 256 f32
            x[c] = tp[c * TILE_PIX + tid];
            if (x[c] > m) { m = x[c]; am = c; }   // '>' keeps first max (argmax)
        }
        float d = 0.0f;
#pragma unroll
        for (int c = 0; c < C_CLS; ++c) { x[c] = __expf(x[c] - m); d += x[c]; }
        const float inv = 1.0f / d;
#pragma unroll
        for (int c = 0; c < C_CLS; ++c) { float p = x[c] * inv; acc[c] += p * p; }
        atomicAdd(&histS[am], 1u);                // ds_add_u32 (exact)
        __syncthreads();                          // done reading before reuse
    }

    // wave32 butterfly reduction of the 19 per-class accumulators
#pragma unroll
    for (int c = 0; c < C_CLS; ++c) {
#pragma unroll
        for (int off = 16; off > 0; off >>= 1)
            acc[c] += __shfl_xor(acc[c], off, 32);
    }
    const int wave = tid >> 5, lane = tid & 31;
    if (lane == 0) {
#pragma unroll
        for (int c = 0; c < C_CLS; ++c) wsum[wave * C_CLS + c] = acc[c];
    }
    __syncthreads();
    if (tid < C_CLS) {
        float s = 0.0f;
#pragma unroll
        for (int w = 0; w < 8; ++w) s += wsum[w * C_CLS + tid];   // fixed order
        Spart[(size_t)blockIdx.x * C_CLS + tid] = s;              // deterministic
        atomicAdd(&gHist[n * C_CLS + tid], histS[tid]);           // integer: exact
    }
}

// ---------------- Kernel 2: weights + deterministic weighted reduction -----
__global__ void __launch_bounds__(256)
iw_final(const float* __restrict__ Spart, const unsigned* __restrict__ gHist,
         float* __restrict__ out) {
    __shared__ float wMat[N_IMG * C_CLS];
    __shared__ double red[8];
    const int tid = threadIdx.x;

    if (tid < N_IMG) {
        float hp[C_CLS];
        float hs = 0.0f;
        for (int c = 0; c < C_CLS; ++c) {
            unsigned h = gHist[tid * C_CLS + c];
            float hf = (h == 0u) ? 1.0f : (float)h;   // hist[hist==0] = 1
            hp[c] = hf;
            hs += hf;                                  // sum AFTER replacement
        }
        for (int c = 0; c < C_CLS; ++c)
            wMat[tid * C_CLS + c] = powf(hs / hp[c], RATIO_F);
    }
    __syncthreads();

    double a = 0.0;
    for (int i = tid; i < GRID * C_CLS; i += 256) {
        int b = i / C_CLS;
        int c = i - b * C_CLS;
        int n = b / BPG;
        a += (double)Spart[i] * (double)wMat[n * C_CLS + c];
    }
    for (int off = 16; off > 0; off >>= 1) a += __shfl_xor(a, off, 32);
    if ((tid & 31) == 0) red[tid >> 5] = a;
    __syncthreads();
    if (tid == 0) {
        double t = 0.0;
#pragma unroll
        for (int w = 0; w < 8; ++w) t += red[w];      // fixed order
        const double total = (double)N_IMG * C_CLS * H_DIM * W_DIM;
        out[0] = (float)(-t / total);
    }
}

extern "C" void kernel_launch(void* const* d_in, const int* in_sizes, int n_in,
                              void* d_out, int out_size, void* d_ws, size_t ws_size,
                              hipStream_t stream) {
    (void)in_sizes; (void)n_in; (void)out_size; (void)ws_size;
    const float* in  = (const float*)d_in[0];
    float*       out = (float*)d_out;
    // Workspace layout: [GRID*C_CLS fp32 partial sums][N_IMG*C_CLS u32 histogram]
    float*    Spart = (float*)d_ws;
    unsigned* gHist = (unsigned*)((char*)d_ws + (size_t)GRID * C_CLS * sizeof(float));

    iw_init <<<1, 256, 0, stream>>>(gHist);
    iw_main <<<GRID, 256, 0, stream>>>(in, Spart, gHist);
    iw_final<<<1, 256, 0, stream>>>(Spart, gHist, out);
}